// VAELoss_18262200943266
// MI455X (gfx1250) — compile-verified
//
#include <hip/hip_runtime.h>
#include <hip/hip_bf16.h>

typedef __attribute__((ext_vector_type(2))) float v2f;
typedef __attribute__((ext_vector_type(8))) float v8f;

#define BS 16
#define NPTS 2048            // N == M == 2048
#define NTILES (NPTS / 16)   // 128 column tiles
#define ROWS_PER_BLOCK 128   // 8 waves * 16 rows
#define THREADS 256

// ---------------------------------------------------------------------------
// Init: colmin[b*M+j] = +inf (as uint bits), rowsum = 0
// ---------------------------------------------------------------------------
__global__ void chamfer_init_kernel(unsigned int* __restrict__ colmin,
                                    float* __restrict__ rowsum) {
    int i = blockIdx.x * blockDim.x + threadIdx.x;
    if (i < BS * NPTS) colmin[i] = 0x7F800000u;  // +inf
    if (i == 0) *rowsum = 0.0f;
}

// ---------------------------------------------------------------------------
// Main fused kernel: one block = 128 rows of p1 for one batch, sweeping all
// 2048 columns of p2 (staged in LDS). Dot products via V_WMMA_F32_16X16X4_F32.
// LDS tile reads are software-pipelined one iteration ahead so the epilogue
// VALU work hides ds latency instead of stalling on s_wait_dscnt.
// ---------------------------------------------------------------------------
__global__ __launch_bounds__(THREADS)
void chamfer_main_kernel(const float* __restrict__ p1,
                         const float* __restrict__ p2,
                         unsigned int* __restrict__ g_colmin,
                         float* __restrict__ g_rowsum) {
    __shared__ float4 s_p2[NPTS];            // (x, y, z, |p|^2)  32 KB
    __shared__ unsigned int s_colmin[NPTS];  // per-block column mins  8 KB

    const int b    = blockIdx.x >> 4;   // 16 row-blocks per batch
    const int rb   = blockIdx.x & 15;
    const int tid  = threadIdx.x;
    const int wave = tid >> 5;
    const int lane = tid & 31;
    const int half = lane >> 4;         // 0: lanes 0-15, 1: lanes 16-31
    const int mr   = lane & 15;

    // ---- stage target cloud for this batch into LDS with precomputed norms
    const float* p2b = p2 + (size_t)b * NPTS * 3;
    for (int j = tid; j < NPTS; j += THREADS) {
        float x = p2b[j * 3 + 0];
        float y = p2b[j * 3 + 1];
        float z = p2b[j * 3 + 2];
        s_p2[j]     = make_float4(x, y, z, x * x + y * y + z * z);
        s_colmin[j] = 0x7F800000u;
    }
    __syncthreads();

    const int row0   = rb * ROWS_PER_BLOCK + wave * 16;
    const float* p1b = p1 + ((size_t)b * NPTS + row0) * 3;

    // ---- A fragment: 16x4 f32, K=3 padded to 4 with zeros
    // lane L: row M = L&15 ; VGPR0 = K(0|2), VGPR1 = K(1|3->0)
    v2f afrag;
    {
        const float* pr = p1b + mr * 3;
        afrag.x = pr[half ? 2 : 0];
        afrag.y = half ? 0.0f : pr[1];
    }

    // ---- squared norms of the 8 rows this lane's half covers (M = v + 8*half)
    float n1v[8];
#pragma unroll
    for (int v = 0; v < 8; ++v) {
        const float* pr = p1b + (v + half * 8) * 3;
        float x = pr[0], y = pr[1], z = pr[2];
        n1v[v] = x * x + y * y + z * z;
    }

    float rowmin[8];
#pragma unroll
    for (int v = 0; v < 8; ++v) rowmin[v] = 3.4e38f;

    // ---- sweep all 128 column tiles, LDS reads pipelined one tile ahead
    float4 q = s_p2[mr];  // tile 0
    for (int t = 0; t < NTILES; ++t) {
        const int col0 = t * 16;
        // prefetch next tile's point (branch-free: last iter re-reads current,
        // keeping EXEC all-ones for the WMMA below)
        const int nidx = (t < NTILES - 1) ? (col0 + 16 + mr) : (col0 + mr);
        float4 qn = s_p2[nidx];

        // B fragment: 4x16 f32, same K striping as A (symmetric layout)
        v2f bfrag;
        bfrag.x = half ? q.z : q.x;
        bfrag.y = half ? 0.0f : q.y;

        v8f c = {};
        c = __builtin_amdgcn_wmma_f32_16x16x4_f32(
            /*neg_a=*/false, afrag, /*neg_b=*/false, bfrag,
            /*c_mod=*/(short)0, c, /*reuse_a=*/false, /*reuse_b=*/false);

        float n2   = q.w;
        float cmin = 3.4e38f;
#pragma unroll
        for (int v = 0; v < 8; ++v) {
            // d^2 = |p1|^2 + |p2|^2 - 2*dot ; clamp tiny negative rounding so
            // uint-bit ordering is valid for the min reductions.
            float d2 = fmaxf(fmaf(-2.0f, c[v], n1v[v] + n2), 0.0f);
            rowmin[v] = fminf(rowmin[v], d2);
            cmin      = fminf(cmin, d2);
        }
        // column-min contribution (both halves cover M=0..15 for this N=mr)
        atomicMin(&s_colmin[col0 + mr], __float_as_uint(cmin));

        q = qn;
    }

    // ---- row-min reduction across the 16 lanes of each half (N dimension)
#pragma unroll
    for (int v = 0; v < 8; ++v) {
        float r = rowmin[v];
        r = fminf(r, __shfl_xor(r, 1, 32));
        r = fminf(r, __shfl_xor(r, 2, 32));
        r = fminf(r, __shfl_xor(r, 4, 32));
        r = fminf(r, __shfl_xor(r, 8, 32));
        rowmin[v] = r;
    }
    if (mr == 0) {  // lane 0 (rows 0-7) and lane 16 (rows 8-15)
        float s = 0.0f;
#pragma unroll
        for (int v = 0; v < 8; ++v) s += rowmin[v];
        atomicAdd(g_rowsum, s);
    }

    // ---- flush block-local column mins to global
    __syncthreads();
    unsigned int* gcm = g_colmin + b * NPTS;
    for (int j = tid; j < NPTS; j += THREADS) {
        atomicMin(&gcm[j], s_colmin[j]);
    }
}

// ---------------------------------------------------------------------------
// Final: sum the 32K column mins, combine with row sum, emit scalar.
// mean over (bs*N) and mean over (bs*M) share denominator 32768.
// ---------------------------------------------------------------------------
__global__ __launch_bounds__(THREADS)
void chamfer_final_kernel(const unsigned int* __restrict__ colmin,
                          const float* __restrict__ rowsum,
                          float* __restrict__ out) {
    __shared__ float red[THREADS];
    float s = 0.0f;
    for (int i = threadIdx.x; i < BS * NPTS; i += THREADS)
        s += __uint_as_float(colmin[i]);
    red[threadIdx.x] = s;
    __syncthreads();
    for (int off = THREADS / 2; off > 0; off >>= 1) {
        if (threadIdx.x < off) red[threadIdx.x] += red[threadIdx.x + off];
        __syncthreads();
    }
    if (threadIdx.x == 0) {
        const float inv = 1.0f / (float)(BS * NPTS);  // 1/32768
        out[0] = (rowsum[0] + red[0]) * inv;          // CD_COEFF == 1
    }
}

// ---------------------------------------------------------------------------
extern "C" void kernel_launch(void* const* d_in, const int* in_sizes, int n_in,
                              void* d_out, int out_size, void* d_ws, size_t ws_size,
                              hipStream_t stream) {
    const float* p1 = (const float*)d_in[0];  // (16, 2048, 3) f32
    const float* p2 = (const float*)d_in[1];  // (16, 2048, 3) f32
    float* out = (float*)d_out;

    // workspace layout: [0, 128KB) colmin uints, then one float rowsum
    unsigned int* colmin = (unsigned int*)d_ws;
    float* rowsum = (float*)((char*)d_ws + (size_t)BS * NPTS * sizeof(unsigned int));

    chamfer_init_kernel<<<(BS * NPTS + THREADS - 1) / THREADS, THREADS, 0, stream>>>(
        colmin, rowsum);

    chamfer_main_kernel<<<BS * (NPTS / ROWS_PER_BLOCK), THREADS, 0, stream>>>(
        p1, p2, colmin, rowsum);

    chamfer_final_kernel<<<1, THREADS, 0, stream>>>(colmin, rowsum, out);
}